// EdgeSelector2WL_52707838656943
// MI455X (gfx1250) — compile-verified
//
#include <hip/hip_runtime.h>

// ============================================================================
// EdgeSelector2WL on MI455X (gfx1250, wave32).
//
// fp32 end-to-end, GEMMs on V_WMMA_F32_16X16X4_F32 (the only fp32 WMMA on
// CDNA5). Roofline: ~75 GFLOP of GEMM vs ~8.6 GB of scatter traffic
// (8 x 2M-edge segment sums x 256B gather + 256B atomic add) -> memory bound
// ~0.4 ms at 23.3 TB/s HBM; fp32 WMMA keeps GEMMs exact and hidden.
// Tile staging uses GLOBAL_LOAD_ASYNC_TO_LDS_B128 (ASYNCcnt) so LDS fills
// overlap the WMMA K-loops without a VGPR round trip.
// ============================================================================

typedef float v2f __attribute__((ext_vector_type(2)));
typedef float v8f __attribute__((ext_vector_type(8)));
typedef int vsi4 __attribute__((vector_size(16)));  // matches builtin param

__device__ __forceinline__ v8f wmma_f32_16x16x4(v2f a, v2f b, v8f c) {
  // (neg_a, A, neg_b, B, c_mod, C, reuse_a, reuse_b)
  return __builtin_amdgcn_wmma_f32_16x16x4_f32(false, a, false, b, (short)0, c,
                                               false, false);
}

#if __has_builtin(__builtin_amdgcn_global_load_async_to_lds_b128)
#define HAVE_ASYNC_LDS 1
#endif

// Copy 16B global -> LDS. Async path: global_load_async_to_lds_b128
// (per-lane VMEM op, tracked by ASYNCcnt, no VGPR round trip).
__device__ __forceinline__ void async_copy_f4(const float* g, float* l) {
#ifdef HAVE_ASYNC_LDS
  __builtin_amdgcn_global_load_async_to_lds_b128(
      (__attribute__((address_space(1))) vsi4*)g,
      (__attribute__((address_space(3))) vsi4*)l, /*offset=*/0, /*cpol=*/0);
#else
  *reinterpret_cast<float4*>(l) = *reinterpret_cast<const float4*>(g);
#endif
}

__device__ __forceinline__ void async_wait() {
#ifdef HAVE_ASYNC_LDS
#if __has_builtin(__builtin_amdgcn_s_wait_asynccnt)
  __builtin_amdgcn_s_wait_asynccnt(0);
#else
  asm volatile("s_wait_asynccnt 0x0" ::: "memory");
#endif
#endif
}

// ---------------------------------------------------------------------------
__global__ void zero_kernel(float* __restrict__ p, int n) {
  int i = blockIdx.x * blockDim.x + threadIdx.x;
  if (i < n) p[i] = 0.0f;
}

// agg[i] = (1+eps) * x[i]  (self term of the GIN segment sum)
__global__ void init_agg_kernel(const float* __restrict__ x, int ldx, int xoff,
                                const float* __restrict__ eps,
                                float* __restrict__ agg, int F, int nrows) {
  int i = blockIdx.x * blockDim.x + threadIdx.x;
  if (i >= nrows * F) return;
  int r = i / F, f = i - r * F;
  agg[i] = (1.0f + eps[0]) * x[(size_t)r * ldx + xoff + f];
}

// agg[dst] += x[src] over E edges, float4 gather + global_atomic_add_f32
__global__ void scatter_add_kernel(const int* __restrict__ ei, int E,
                                   const float* __restrict__ x, int ldx,
                                   int xoff, float* __restrict__ agg, int F) {
  int chunks = F >> 2;
  int gid = blockIdx.x * blockDim.x + threadIdx.x;
  if (gid >= E * chunks) return;
  int e = gid / chunks;
  int c = (gid - e * chunks) << 2;
  int src = ei[e];
  int dst = ei[E + e];
  const float4 v =
      *reinterpret_cast<const float4*>(x + (size_t)src * ldx + xoff + c);
  float* ad = agg + (size_t)dst * F + c;
  atomicAdd(ad + 0, v.x);
  atomicAdd(ad + 1, v.y);
  atomicAdd(ad + 2, v.z);
  atomicAdd(ad + 3, v.w);
}

// ---------------------------------------------------------------------------
// Fused GIN MLP: Y[:, yoff:yoff+64] = relu(relu(U @ Wa + ba) @ Wb + bb)
// U: [nrows, KIN] contiguous. 32-row tiles, 8 waves, one 16x16 WMMA tile/wave.
template <int KIN>
__launch_bounds__(256) __global__
void gin_mlp_kernel(const float* __restrict__ U, const float* __restrict__ Wa,
                    const float* __restrict__ ba, const float* __restrict__ Wb,
                    const float* __restrict__ bb, float* __restrict__ Y,
                    int ldy, int yoff, int nrows) {
  __shared__ float sU[32][KIN + 4];  // +4 pad: 16B-multiple row stride
  __shared__ float sWa[KIN][64];
  __shared__ float sZ[32][68];
  __shared__ float sWb[64][64];

  const int tid = threadIdx.x;
  // weights: contiguous async B128 streams
  for (int i = tid; i < KIN * 16; i += 256)
    async_copy_f4(Wa + i * 4, &sWa[0][0] + i * 4);
  for (int i = tid; i < 64 * 16; i += 256)
    async_copy_f4(Wb + i * 4, &sWb[0][0] + i * 4);

  const int wave = tid >> 5, lane = tid & 31;
  const int half = lane >> 4, l15 = lane & 15;
  const int trow = (wave >> 2) << 4;  // 0 / 16
  const int tcol = (wave & 3) << 4;   // 0..48

  const int ntiles = nrows >> 5;
  for (int tile = blockIdx.x; tile < ntiles; tile += gridDim.x) {
    const int row0 = tile << 5;
    __syncthreads();  // prior iteration fully consumed LDS
    constexpr int CH = (32 * KIN) >> 2;
    for (int i = tid; i < CH; i += 256) {
      int r = (i << 2) / KIN;
      int c = (i << 2) % KIN;
      async_copy_f4(&U[(size_t)(row0 + r) * KIN + c], &sU[r][c]);
    }
    async_wait();
    __syncthreads();

    // GEMM1: Z = relu(U @ Wa + ba)
    v8f acc = {};
    for (int k = 0; k < KIN; k += 4) {
      v2f a, b;
      // A 16x4 f32: lanes0-15 K={k,k+1}, lanes16-31 K={k+2,k+3}
      a.x = sU[trow + l15][k + 2 * half];
      a.y = sU[trow + l15][k + 2 * half + 1];
      // B 4x16 f32: row striped across lanes per VGPR
      b.x = sWa[k + 2 * half][tcol + l15];
      b.y = sWa[k + 2 * half + 1][tcol + l15];
      acc = wmma_f32_16x16x4(a, b, acc);
    }
    const float bia = ba[tcol + l15];
#pragma unroll
    for (int r = 0; r < 8; ++r) {
      // D vgpr r: lanes0-15 -> M=r, lanes16-31 -> M=r+8
      sZ[trow + r + 8 * half][tcol + l15] = fmaxf(acc[r] + bia, 0.0f);
    }
    __syncthreads();

    // GEMM2: Y = relu(Z @ Wb + bb)
    v8f acc2 = {};
    for (int k = 0; k < 64; k += 4) {
      v2f a, b;
      a.x = sZ[trow + l15][k + 2 * half];
      a.y = sZ[trow + l15][k + 2 * half + 1];
      b.x = sWb[k + 2 * half][tcol + l15];
      b.y = sWb[k + 2 * half + 1][tcol + l15];
      acc2 = wmma_f32_16x16x4(a, b, acc2);
    }
    const float bib = bb[tcol + l15];
#pragma unroll
    for (int r = 0; r < 8; ++r) {
      float v = fmaxf(acc2[r] + bib, 0.0f);
      Y[(size_t)(row0 + trow + r + 8 * half) * ldy + yoff + tcol + l15] = v;
    }
  }
}

// ---------------------------------------------------------------------------
// Y = X[:, xoff:xoff+K] @ W + bias (no activation; BN stats accumulated).
// K chunked by 64 so LDS stays ~26 KB for K up to 256.
template <int K>
__launch_bounds__(256) __global__
void gemm_stats_kernel(const float* __restrict__ X, int ldx, int xoff,
                       const float* __restrict__ W,
                       const float* __restrict__ bias, float* __restrict__ Y,
                       float* __restrict__ gsum, float* __restrict__ gsq,
                       int nrows) {
  __shared__ float sX[32][68];
  __shared__ float sW[64][64];
  __shared__ float sSum[64];
  __shared__ float sSq[64];

  const int tid = threadIdx.x;
  if (tid < 64) {
    sSum[tid] = 0.0f;
    sSq[tid] = 0.0f;
  }
  const int wave = tid >> 5, lane = tid & 31;
  const int half = lane >> 4, l15 = lane & 15;
  const int trow = (wave >> 2) << 4;
  const int tcol = (wave & 3) << 4;

  float csum = 0.0f, csq = 0.0f;
  const int ntiles = nrows >> 5;
  for (int tile = blockIdx.x; tile < ntiles; tile += gridDim.x) {
    const int row0 = tile << 5;
    v8f acc = {};
    for (int k0 = 0; k0 < K; k0 += 64) {
      __syncthreads();
      // X tile: 32x64, 16 B128 chunks per row
      for (int i = tid; i < 32 * 16; i += 256) {
        int r = i >> 4, c = (i & 15) << 2;
        async_copy_f4(&X[(size_t)(row0 + r) * ldx + xoff + k0 + c], &sX[r][c]);
      }
      // W chunk: contiguous 64x64
      for (int i = tid; i < 64 * 16; i += 256) {
        async_copy_f4(W + (size_t)k0 * 64 + i * 4, &sW[0][0] + i * 4);
      }
      async_wait();
      __syncthreads();
      for (int k = 0; k < 64; k += 4) {
        v2f a, b;
        a.x = sX[trow + l15][k + 2 * half];
        a.y = sX[trow + l15][k + 2 * half + 1];
        b.x = sW[k + 2 * half][tcol + l15];
        b.y = sW[k + 2 * half + 1][tcol + l15];
        acc = wmma_f32_16x16x4(a, b, acc);
      }
    }
    const float bi = bias[tcol + l15];
#pragma unroll
    for (int r = 0; r < 8; ++r) {
      float v = acc[r] + bi;
      Y[(size_t)(row0 + trow + r + 8 * half) * 64 + tcol + l15] = v;
      csum += v;
      csq += v * v;
    }
  }
  __syncthreads();
  atomicAdd(&sSum[tcol + l15], csum);  // ds_add_f32
  atomicAdd(&sSq[tcol + l15], csq);
  __syncthreads();
  if (tid < 64) {
    atomicAdd(&gsum[tid], sSum[tid]);
    atomicAdd(&gsq[tid], sSq[tid]);
  }
}

// ---------------------------------------------------------------------------
// Y[:, yoff:yoff+64] = bn(P) (optional relu). Population var = E[x^2]-E[x]^2.
__global__ void bn_apply_kernel(const float* __restrict__ P,
                                const float* __restrict__ gsum,
                                const float* __restrict__ gsq,
                                const float* __restrict__ g,
                                const float* __restrict__ b,
                                float* __restrict__ Y, int ldy, int yoff,
                                int relu, int nrows) {
  int i = blockIdx.x * blockDim.x + threadIdx.x;
  if (i >= nrows * 64) return;
  int r = i >> 6, c = i & 63;
  float inv = 1.0f / (float)nrows;
  float m = gsum[c] * inv;
  float var = gsq[c] * inv - m * m;
  float v = g[c] * (P[i] - m) * rsqrtf(var + 1e-5f) + b[c];
  if (relu) v = fmaxf(v, 0.0f);
  Y[(size_t)r * ldy + yoff + c] = v;
}

// scores[r] = dot(Z[r,:64], W1) + b1   (W1 is [64,1])
__global__ void head_dot_kernel(const float* __restrict__ Z,
                                const float* __restrict__ W1,
                                const float* __restrict__ b1,
                                float* __restrict__ scores, int nrows) {
  int r = blockIdx.x * blockDim.x + threadIdx.x;
  if (r >= nrows) return;
  const float* z = Z + (size_t)r * 64;
  float s = b1[0];
#pragma unroll
  for (int f = 0; f < 64; ++f) s += z[f] * W1[f];
  scores[r] = s;
}

// Final gathers. out layout: [0,C) select | [C,C+E) delete | [C+E, C+E+2C) idx
__global__ void gather_kernel(const int* __restrict__ cand,
                              const int* __restrict__ ei,
                              const int* __restrict__ nnodes,
                              const float* __restrict__ selS,
                              const float* __restrict__ delS,
                              float* __restrict__ out, int C, int E,
                              int candPer, int edgesPer) {
  int t = blockIdx.x * blockDim.x + threadIdx.x;
  if (t < C) {
    int g = t / candPer;
    int nn = nnodes[g];
    int i0 = cand[2 * t], i1 = cand[2 * t + 1];
    size_t cs2 = (size_t)g * nn * nn;
    out[t] = selS[cs2 + (size_t)i0 * nn + i1];
    out[(size_t)C + E + 2 * (size_t)t + 0] = (float)(i0 + g * nn);
    out[(size_t)C + E + 2 * (size_t)t + 1] = (float)(i1 + g * nn);
  }
  if (t < E) {
    int g = t / edgesPer;
    int nn = nnodes[g];
    int inc = g * nn;
    int l0 = ei[t] - inc, l1 = ei[E + t] - inc;
    out[C + t] = delS[(size_t)g * nn * nn + (size_t)l0 * nn + l1];
  }
}

// ===========================================================================
extern "C" void kernel_launch(void* const* d_in, const int* in_sizes, int n_in,
                              void* d_out, int out_size, void* d_ws,
                              size_t ws_size, hipStream_t stream) {
  const float* x2wl = (const float*)d_in[0];
  const int* ei1 = (const int*)d_in[1];
  const int* ei2 = (const int*)d_in[2];
  const int* ei = (const int*)d_in[3];
  const int* cand = (const int*)d_in[4];
  const int* nnodes = (const int*)d_in[5];

  const int nrows = in_sizes[0] / 16;  // N2 = 262144
  const int E1 = in_sizes[1] / 2;
  const int E2 = in_sizes[2] / 2;
  const int Eg = in_sizes[3] / 2;
  const int Cg = in_sizes[4] / 2;
  const int Bg = in_sizes[5];
  const int candPer = Cg / Bg;
  const int edgesPer = Eg / Bg;

  // Params: JAX pytree flatten = dict keys ASCII-sorted, lists in order.
  // Per conv layer (18): 0 bnb,1 bng,2 c1_Wa,3 c1_Wb,4 c1_ba,5 c1_bb,6 c1_eps,
  //   7 c2_Wa,8 c2_Wb,9 c2_ba,10 c2_bb,11 c2_eps,
  //   12 imWa,13 imWb,14 imb,15 imba,16 imbb,17 img
  // Heads at 8+72 (6 each): 0 W0,1 W1,2 b0,3 b1,4 bt0,5 g0
  const int P0 = 8;
  auto prm = [&](int idx) { return (const float*)d_in[P0 + idx]; };

  float* ws = (float*)d_ws;
  const size_t N2s = (size_t)nrows;
  float* inter = ws;                // [N2, 256] layer outputs (xc)
  float* buf1 = inter + N2s * 256;  // [N2, 64]
  float* buf2 = buf1 + N2s * 64;    // [N2, 64]
  float* buf3 = buf2 + N2s * 64;    // [N2, 128] h12 / scores
  float* stats = buf3 + N2s * 128;  // 10 slots x 128 (sum|sumsq)

  auto gsz = [](long long n) { return (unsigned)((n + 255) / 256); };

  zero_kernel<<<gsz(10 * 128), 256, 0, stream>>>(stats, 10 * 128);

  for (int L = 0; L < 4; ++L) {
    const int lo = 18 * L;
    const float* xsrc;
    int ldx, xoff, KIN;
    if (L == 0) {
      xsrc = x2wl; ldx = 16; xoff = 0; KIN = 16;
    } else {
      xsrc = inter; ldx = 256; xoff = 64 * (L - 1); KIN = 64;
    }

    // agg = (1+eps)*x, then scatter-add neighbors
    init_agg_kernel<<<gsz((long long)nrows * KIN), 256, 0, stream>>>(
        xsrc, ldx, xoff, prm(lo + 6), buf1, KIN, nrows);
    init_agg_kernel<<<gsz((long long)nrows * KIN), 256, 0, stream>>>(
        xsrc, ldx, xoff, prm(lo + 11), buf2, KIN, nrows);
    const int ch = KIN >> 2;
    scatter_add_kernel<<<gsz((long long)E1 * ch), 256, 0, stream>>>(
        ei1, E1, xsrc, ldx, xoff, buf1, KIN);
    scatter_add_kernel<<<gsz((long long)E2 * ch), 256, 0, stream>>>(
        ei2, E2, xsrc, ldx, xoff, buf2, KIN);

    // x1|x2 -> buf3 [N2,128]
    int gb = nrows / 32 < 8192 ? nrows / 32 : 8192;
    if (KIN == 16) {
      gin_mlp_kernel<16><<<gb, 256, 0, stream>>>(buf1, prm(lo + 2), prm(lo + 4),
                                                 prm(lo + 3), prm(lo + 5), buf3,
                                                 128, 0, nrows);
      gin_mlp_kernel<16><<<gb, 256, 0, stream>>>(buf2, prm(lo + 7), prm(lo + 9),
                                                 prm(lo + 8), prm(lo + 10),
                                                 buf3, 128, 64, nrows);
    } else {
      gin_mlp_kernel<64><<<gb, 256, 0, stream>>>(buf1, prm(lo + 2), prm(lo + 4),
                                                 prm(lo + 3), prm(lo + 5), buf3,
                                                 128, 0, nrows);
      gin_mlp_kernel<64><<<gb, 256, 0, stream>>>(buf2, prm(lo + 7), prm(lo + 9),
                                                 prm(lo + 8), prm(lo + 10),
                                                 buf3, 128, 64, nrows);
    }

    // inter-MLP: relu(bn(h12 @ imWa + imba)) @ imWb + imbb, then bn -> inter
    float* sA = stats + (size_t)(2 * L) * 128;
    float* sB = stats + (size_t)(2 * L + 1) * 128;
    gemm_stats_kernel<128><<<2048, 256, 0, stream>>>(
        buf3, 128, 0, prm(lo + 12), prm(lo + 15), buf1, sA, sA + 64, nrows);
    bn_apply_kernel<<<gsz((long long)nrows * 64), 256, 0, stream>>>(
        buf1, sA, sA + 64, prm(lo + 17), prm(lo + 14), buf2, 64, 0, 1, nrows);
    gemm_stats_kernel<64><<<2048, 256, 0, stream>>>(
        buf2, 64, 0, prm(lo + 13), prm(lo + 16), buf1, sB, sB + 64, nrows);
    bn_apply_kernel<<<gsz((long long)nrows * 64), 256, 0, stream>>>(
        buf1, sB, sB + 64, prm(lo + 1), prm(lo + 0), inter, 256, 64 * L, 0,
        nrows);
  }

  // Heads: scores = (relu(bn(xc @ W0 + b0))) @ W1 + b1
  for (int h = 0; h < 2; ++h) {
    const int ho = 80 + 6 * h;  // 8 + 4*18 = 80
    float* sH = stats + (size_t)(8 + h) * 128;
    gemm_stats_kernel<256><<<2048, 256, 0, stream>>>(
        inter, 256, 0, prm(ho + 0), prm(ho + 2), buf1, sH, sH + 64, nrows);
    bn_apply_kernel<<<gsz((long long)nrows * 64), 256, 0, stream>>>(
        buf1, sH, sH + 64, prm(ho + 5), prm(ho + 4), buf2, 64, 0, 1, nrows);
    head_dot_kernel<<<gsz(nrows), 256, 0, stream>>>(
        buf2, prm(ho + 1), prm(ho + 3), buf3 + (size_t)h * nrows, nrows);
  }

  // Gathers -> d_out (select | delete | edge_candidate_idx as floats)
  float* out = (float*)d_out;
  int tmax = Cg > Eg ? Cg : Eg;
  gather_kernel<<<gsz(tmax), 256, 0, stream>>>(cand, ei, nnodes, buf3,
                                               buf3 + N2s, out, Cg, Eg, candPer,
                                               edgesPer);
}